// CorticalGrid_16930761081556
// MI455X (gfx1250) — compile-verified
//
#include <hip/hip_runtime.h>
#include <math.h>

typedef __attribute__((ext_vector_type(2))) float v2f;
typedef __attribute__((ext_vector_type(8))) float v8f;

#define GH 64
#define GW 64
#define PH 3
#define PW 3
#define OBJ 16
#define LOC 4
#define SDIM 9          // sensory dim
#define DDIM 20         // obj+loc
#define NCOL 4096       // columns
#define BATCH 1024
#define LAMBDA 0.1f
#define ETA 0.05f
#define NSTEPS 10

#define NP ((long long)NCOL * BATCH * SDIM)   // patches elements
#define NX ((long long)NCOL * BATCH * DDIM)   // state elements

// ---------------------------------------------------------------------------
// Hardware tanh (V_TANH_F32 is a CDNA5 TRANS op). Fallback is branch-free.
// ---------------------------------------------------------------------------
__device__ __forceinline__ float fast_tanh(float x) {
#if __has_builtin(__builtin_amdgcn_tanhf)
    return __builtin_amdgcn_tanhf(x);
#elif __has_builtin(__builtin_amdgcn_tanh_f32)
    return __builtin_amdgcn_tanh_f32(x);
#else
    float t = __expf(-2.0f * fabsf(x));          // v_exp_f32, branch-free
    float r = (1.0f - t) * __frcp_rn(1.0f + t);
    return copysignf(r, x);
#endif
}

// ---------------------------------------------------------------------------
// Rearrange global_input (B, GH*PH*GW*PW) -> patches (N, B, S), contiguous in s.
// One-time cost; after this, patches (151 MB) is L2-resident (192 MB L2).
// ---------------------------------------------------------------------------
__global__ void cg_patch_kernel(const float* __restrict__ gin,
                                float* __restrict__ patches) {
    long long idx    = (long long)blockIdx.x * blockDim.x + threadIdx.x;
    long long stride = (long long)gridDim.x * blockDim.x;
    for (; idx < NP; idx += stride) {
        int s = (int)(idx % SDIM);
        long long t = idx / SDIM;
        int b = (int)(t % BATCH);
        int n = (int)(t / BATCH);
        int r = n / GW, c = n % GW;
        int pr = s / PW, pc = s % PW;
        long long in = (long long)b * (GH * PH * GW * PW)
                     + (long long)r * (PH * GW * PW)
                     + (long long)pr * (GW * PW)
                     + (long long)c * PW + pc;
        patches[idx] = gin[in];
    }
}

__global__ void cg_zero_kernel(float* __restrict__ p, long long count) {
    long long idx    = (long long)blockIdx.x * blockDim.x + threadIdx.x;
    long long stride = (long long)gridDim.x * blockDim.x;
    for (; idx < count; idx += stride) p[idx] = 0.0f;
}

// ---------------------------------------------------------------------------
// One inference step. One block per column n; 8 waves; each wave owns 8
// batch-tiles of 16 rows.  All GEMMs via V_WMMA_F32_16X16X4_F32 (f32 exact,
// memory-bound problem so the f32 matrix rate is free).
//
// Layouts (wave32):
//   A 16x4 f32 : lane<16 -> M=lane, K={0,1} in {v0,v1}; lane>=16 -> K={2,3}
//   B 4x16 f32 : lane<16 -> N=lane, K={0,1};            lane>=16 -> K={2,3}
//   C 16x16 f32: comp v, lane l: M = v + 8*(l>>4), N = l&15
// ---------------------------------------------------------------------------
__global__ __launch_bounds__(256) void cg_step_kernel(
    const float* __restrict__ x_in,   // (N, B, D)  step-start state
    float* __restrict__ x_out,        // (N, B, D)  step-end state
    const float* __restrict__ patches,// (N, B, S)
    const float* __restrict__ Wg,     // (N, S, D)
    const int*  __restrict__ nbr,     // (N, 4)
    float* __restrict__ energy_part,  // (NSTEPS, NCOL) block partial energies
    int step)
{
    __shared__ float Wlds[SDIM * DDIM];     // 180 floats, per-column weights
    __shared__ float tlds[8][16][16];       // per-wave t-tile transpose buffer
    __shared__ float red[256];

    const int n    = blockIdx.x;
    const int tid  = threadIdx.x;
    const int wave = tid >> 5;
    const int lane = tid & 31;
    const int hi   = lane >> 4;      // which half of the wave
    const int lr   = lane & 15;

    // stage W_n into LDS
    for (int i = tid; i < SDIM * DDIM; i += 256) Wlds[i] = Wg[(long long)n * SDIM * DDIM + i];
    __syncthreads();

    // neighbor bookkeeping (scalar per block, uniform across lanes)
    int   nb[4];
    float nmask[4];
    float cnt = 0.0f;
    #pragma unroll
    for (int j = 0; j < 4; ++j) {
        int v = nbr[n * 4 + j];
        nb[j]    = v < 0 ? 0 : v;
        nmask[j] = v < 0 ? 0.0f : 1.0f;
        cnt += nmask[j];
    }
    if (cnt < 1.0f) cnt = 1.0f;
    const float inv_cnt = 1.0f / cnt;

    const float smaskf  = (lr < SDIM) ? 1.0f : 0.0f;   // s-column validity
    const int   sclamp  = (lr < SDIM) ? lr : 0;        // safe patch column

    // B-operand for pred:  Wt (D x S): element (K=d, N=s) = W[s][d]; pad s>=9 with 0
    v2f b1[5];
    #pragma unroll
    for (int kc = 0; kc < 5; ++kc) {
        int d0 = kc * 4 + 2 * hi;
        float e0 = smaskf * Wlds[sclamp * DDIM + d0];
        float e1 = smaskf * Wlds[sclamp * DDIM + d0 + 1];
        v2f t = {e0, e1};
        b1[kc] = t;
    }
    // B-operand for g:  W (S x D): element (K=s, N=d) = W[s][d]; pad s>=9 / d>=20
    v2f b2[2][3];
    #pragma unroll
    for (int nt = 0; nt < 2; ++nt) {
        int d = nt * 16 + lr;
        int dc = d < DDIM ? d : 0;
        float dm = d < DDIM ? 1.0f : 0.0f;
        #pragma unroll
        for (int kc = 0; kc < 3; ++kc) {
            int s0 = kc * 4 + 2 * hi;
            float e0 = (s0     < SDIM) ? dm * Wlds[s0 * DDIM + dc]       : 0.0f;
            float e1 = (s0 + 1 < SDIM) ? dm * Wlds[(s0 + 1) * DDIM + dc] : 0.0f;
            v2f t = {e0, e1};
            b2[nt][kc] = t;
        }
    }

    float elocal = 0.0f;

    for (int t = 0; t < 8; ++t) {
        const int bt = (wave + 8 * t) * 16;   // batch-tile base row

        // ---- pred = tanh(x . W^T): A = x tile (16 x 20), 5 K-chunks --------
        const float* xrow = x_in + ((long long)n * BATCH + bt + lr) * DDIM;
        // speculative prefetch of next tile's row (safe past the end)
        __builtin_prefetch(xrow + 128 * DDIM, 0, 3);
        v8f acc = {};
        #pragma unroll
        for (int kc = 0; kc < 5; ++kc) {
            v2f a = *(const v2f*)(xrow + kc * 4 + 2 * hi);   // 8B aligned (even index)
            acc = __builtin_amdgcn_wmma_f32_16x16x4_f32(
                false, a, false, b1[kc], (short)0, acc, false, false);
        }

        // ---- eps, t = eps*(1-pred^2); padded s-cols are exactly 0 ----------
        const float* prow = patches + ((long long)n * BATCH + bt) * SDIM + sclamp;
        #pragma unroll
        for (int v = 0; v < 8; ++v) {
            int m = v + 8 * hi;                               // batch row in tile
            float p   = smaskf * prow[(long long)m * SDIM];   // unconditional load + mask
            float pd  = fast_tanh(acc[v]);
            float eps = p - pd;
            elocal += 0.5f * eps * eps;
            tlds[wave][m][lr] = eps * (1.0f - pd * pd);
        }
        asm volatile("s_wait_dscnt 0" ::: "memory");          // wave-local LDS transpose

        // t in A layout (M=batch, K=s), 3 K-chunks (s padded 9->12, pads are 0)
        v2f a2[3];
        #pragma unroll
        for (int kc = 0; kc < 3; ++kc) {
            int s0 = kc * 4 + 2 * hi;
            a2[kc] = *(const v2f*)&tlds[wave][lr][s0];
        }

        // ---- g = t . W, two N-tiles (d 0..15 = OBJ, 16..19 = LOC) ----------
        #pragma unroll
        for (int nt = 0; nt < 2; ++nt) {
            v8f g = {};
            #pragma unroll
            for (int kc = 0; kc < 3; ++kc) {
                g = __builtin_amdgcn_wmma_f32_16x16x4_f32(
                    false, a2[kc], false, b2[nt][kc], (short)0, g, false, false);
            }
            int d = nt * 16 + lr;
            if (d < DDIM) {
                #pragma unroll
                for (int v = 0; v < 8; ++v) {
                    int m = v + 8 * hi;
                    long long bi = (long long)n * BATCH + bt + m;
                    float xo = x_in[bi * DDIM + d];
                    float xn;
                    if (nt == 0) {  // OBJ dims: lateral coupling + energy
                        float ctx = 0.0f;
                        #pragma unroll
                        for (int j = 0; j < 4; ++j)
                            ctx += nmask[j] *
                                x_in[((long long)nb[j] * BATCH + bt + m) * DDIM + d];
                        ctx *= inv_cnt;
                        float el = xo - ctx;
                        elocal += 0.5f * LAMBDA * el * el;
                        xn = xo + ETA * (g[v] - LAMBDA * el);
                    } else {        // LOC dims
                        xn = xo + ETA * g[v];
                    }
                    x_out[bi * DDIM + d] = xn;
                }
            }
        }
    }

    // deterministic block reduction of energy
    red[tid] = elocal;
    __syncthreads();
    #pragma unroll
    for (int off = 128; off > 0; off >>= 1) {
        if (tid < off) red[tid] += red[tid + off];
        __syncthreads();
    }
    if (tid == 0) energy_part[(long long)step * NCOL + blockIdx.x] = red[0];
}

// one block per step: deterministic tree-sum of 4096 partials
__global__ void cg_reduce_kernel(const float* __restrict__ partial,
                                 float* __restrict__ out) {
    __shared__ float red[256];
    int step = blockIdx.x;
    float s = 0.0f;
    for (int i = threadIdx.x; i < NCOL; i += 256)
        s += partial[(long long)step * NCOL + i];
    red[threadIdx.x] = s;
    __syncthreads();
    #pragma unroll
    for (int off = 128; off > 0; off >>= 1) {
        if (threadIdx.x < off) red[threadIdx.x] += red[threadIdx.x + off];
        __syncthreads();
    }
    if (threadIdx.x == 0) out[step] = red[0];
}

extern "C" void kernel_launch(void* const* d_in, const int* in_sizes, int n_in,
                              void* d_out, int out_size, void* d_ws, size_t ws_size,
                              hipStream_t stream) {
    (void)in_sizes; (void)n_in; (void)out_size; (void)ws_size;
    const float* gin = (const float*)d_in[0];   // (B, GH*PH*GW*PW)
    const float* Wg  = (const float*)d_in[1];   // (N, S, D)
    const int*   nbr = (const int*)d_in[2];     // (N, 4)
    // d_in[3] = steps (==NSTEPS, fixed at compile time)

    float* ws      = (float*)d_ws;
    float* patches = ws;                 // NP floats
    float* xA      = patches + NP;       // NX floats (state, double buffered)
    float* xB      = xA + NX;            // NX floats
    float* epart   = xB + NX;            // NSTEPS*NCOL floats

    cg_patch_kernel<<<4096, 256, 0, stream>>>(gin, patches);
    cg_zero_kernel <<<4096, 256, 0, stream>>>(xA, NX);       // x0 = 0

    float* xi = xA;
    float* xo = xB;
    for (int s = 0; s < NSTEPS; ++s) {
        cg_step_kernel<<<NCOL, 256, 0, stream>>>(xi, xo, patches, Wg, nbr, epart, s);
        float* tmp = xi; xi = xo; xo = tmp;
    }
    cg_reduce_kernel<<<NSTEPS, 256, 0, stream>>>(epart, (float*)d_out);
}